// DecDTWLayer_33406255628873
// MI455X (gfx1250) — compile-verified
//
#include <hip/hip_runtime.h>

// Problem constants (match reference)
#define B_    32
#define N1_   512
#define N2_   512
#define D_    128
#define MW    256     // M_WARP
#define KD    96      // M_DISCR
#define GAMMA_    0.1f
#define INV_GAMMA 10.0f
#define BARRIER_  10000.0f

typedef float v2f __attribute__((ext_vector_type(2)));
typedef float v8f __attribute__((ext_vector_type(8)));

// ---------------------------------------------------------------------------
// Kernel 1: node matrix via f32 WMMA Gram matrix.
//   node[b,m,k] = wts[b,m] * ( ||s1_at||^2 + ||s2_at||^2 - 2*s1_at.s2_at )
//   with s2_at = (1-w)*f[i0] + w*f[i0+1]  =>  dot terms are gathers from
//   G[m,n] = s1_at[m] . s2f[n]  (WMMA GEMM), nrm2[n], adj[n]=f[n].f[n+1].
// Grid: (MW/16, B), Block: 256 (8 waves; each wave owns 16x16 G tiles)
// ---------------------------------------------------------------------------
__global__ __launch_bounds__(256) void gdtw_node_kernel(
    const float* __restrict__ s1f, const float* __restrict__ s2f,
    const float* __restrict__ glbL, const float* __restrict__ glbU,
    const float* __restrict__ t1,  const float* __restrict__ t2,
    const float* __restrict__ tw,  float* __restrict__ node_out)
{
  const int b   = blockIdx.y;
  const int m0  = blockIdx.x * 16;
  const int tid = threadIdx.x;

  __shared__ float s1_lds[16][D_];   // interpolated s1 tile (A matrix)
  __shared__ float G[16][N2_];       // Gram tile, 16 x 512
  __shared__ float nrm2[N2_];
  __shared__ float adj[N2_];
  __shared__ float nrm1[16];
  __shared__ float red[256];
  __shared__ float Tmax1, Tmax2;

  // --- T1 = max(t1[b,:]), T2 = max(t2[b,:]) ---
  {
    float m1 = -1e30f, m2 = -1e30f;
    for (int i = tid; i < N1_; i += 256) m1 = fmaxf(m1, t1[b * N1_ + i]);
    for (int i = tid; i < N2_; i += 256) m2 = fmaxf(m2, t2[b * N2_ + i]);
    red[tid] = m1; __syncthreads();
    for (int s = 128; s > 0; s >>= 1) { if (tid < s) red[tid] = fmaxf(red[tid], red[tid + s]); __syncthreads(); }
    if (tid == 0) Tmax1 = red[0];
    __syncthreads();
    red[tid] = m2; __syncthreads();
    for (int s = 128; s > 0; s >>= 1) { if (tid < s) red[tid] = fmaxf(red[tid], red[tid + s]); __syncthreads(); }
    if (tid == 0) Tmax2 = red[0];
    __syncthreads();
  }
  const float T1 = Tmax1, T2 = Tmax2;

  // --- s1_at tile: lerp of s1f at tw/T1 ---
  for (int idx = tid; idx < 16 * D_; idx += 256) {
    const int r = idx >> 7, d = idx & (D_ - 1);
    const int m = m0 + r;
    float pos = tw[b * MW + m] / T1;
    pos = fminf(fmaxf(pos, 0.f), 1.f);
    const float x = pos * (float)(N1_ - 1);
    int i0 = (int)x; i0 = (i0 < 0) ? 0 : ((i0 > N1_ - 2) ? N1_ - 2 : i0);
    const float w = x - (float)i0;
    const float* base = s1f + ((size_t)b * N1_ + i0) * D_ + d;
    s1_lds[r][d] = base[0] * (1.f - w) + base[D_] * w;
  }
  __syncthreads();

  if (tid < 16) {
    float s = 0.f;
    for (int d = 0; d < D_; ++d) { const float v = s1_lds[tid][d]; s += v * v; }
    nrm1[tid] = s;
  }
  // --- per-batch row norms / adjacent dots of s2f ---
  for (int n = tid; n < N2_; n += 256) {
    const float* f = s2f + ((size_t)b * N2_ + n) * D_;
    float s = 0.f, sa = 0.f;
    if (n < N2_ - 1) {
      for (int d = 0; d < D_; ++d) { const float v = f[d]; s += v * v; sa += v * f[D_ + d]; }
    } else {
      for (int d = 0; d < D_; ++d) { const float v = f[d]; s += v * v; }
    }
    nrm2[n] = s; adj[n] = (n < N2_ - 1) ? sa : 0.f;
  }
  __syncthreads();

  // --- WMMA GEMM: G[16 x 512] = s1_tile[16 x 128] * s2f^T[128 x 512] ---
  // f32 WMMA 16x16x4 layouts (ISA 7.12.2):
  //   A: lane%16 = M; VGPR0 = K(2*(lane/16)),   VGPR1 = K(2*(lane/16)+1)
  //   B: lane%16 = N; VGPR0 = K(2*(lane/16)),   VGPR1 = K(2*(lane/16)+1)
  //   D: col = lane%16; VGPR r -> row r + 8*(lane/16)
  {
    const int wave = tid >> 5;
    const int lane = tid & 31;
    const int lrow = lane & 15;
    const int lhi  = lane >> 4;
    const float* s2base = s2f + (size_t)b * N2_ * D_;
    for (int nt = wave; nt < N2_ / 16; nt += 8) {
      const int n0 = nt * 16;
      v8f acc = {0.f, 0.f, 0.f, 0.f, 0.f, 0.f, 0.f, 0.f};
      for (int kk = 0; kk < D_; kk += 4) {
        v2f a, bm;
        a.x = s1_lds[lrow][kk + 2 * lhi + 0];
        a.y = s1_lds[lrow][kk + 2 * lhi + 1];
        const float* brow = s2base + (size_t)(n0 + lrow) * D_ + kk + 2 * lhi;
        bm.x = brow[0];
        bm.y = brow[1];
        acc = __builtin_amdgcn_wmma_f32_16x16x4_f32(
            /*neg_a=*/false, a, /*neg_b=*/false, bm,
            /*c_mod=*/(short)0, acc, /*reuse_a=*/false, /*reuse_b=*/false);
      }
#pragma unroll
      for (int r = 0; r < 8; ++r) G[r + 8 * lhi][n0 + lrow] = acc[r];
    }
  }
  __syncthreads();

  // --- assemble node tile and store ---
  for (int idx = tid; idx < 16 * KD; idx += 256) {
    const int r = idx / KD, k = idx - r * KD;
    const int m = m0 + r;
    const float lb = glbL[b * MW + m] * T2;
    const float ub = glbU[b * MW + m] * T2;
    const float tau = lb + (ub - lb) * ((float)k * (1.f / (float)(KD - 1)));
    float pos = fminf(fmaxf(tau / T2, 0.f), 1.f);
    const float x = pos * (float)(N2_ - 1);
    int i0 = (int)x; i0 = (i0 < 0) ? 0 : ((i0 > N2_ - 2) ? N2_ - 2 : i0);
    const float w = x - (float)i0;
    const float S  = (1.f - w) * G[r][i0] + w * G[r][i0 + 1];
    const float q2 = (1.f - w) * (1.f - w) * nrm2[i0]
                   + 2.f * w * (1.f - w) * adj[i0]
                   + w * w * nrm2[i0 + 1];
    const float d2 = nrm1[r] + q2 - 2.f * S;
    const float wl = (m > 0)      ? (tw[b * MW + m] - tw[b * MW + m - 1]) : 0.f;
    const float wr = (m < MW - 1) ? (tw[b * MW + m + 1] - tw[b * MW + m]) : 0.f;
    float nodev = d2 * 0.5f * (wl + wr);
    if (m == 0)      nodev += BARRIER_ * tau * tau;
    if (m == MW - 1) { const float dd = tau - T2; nodev += BARRIER_ * dd * dd; }
    node_out[((size_t)b * MW + m) * KD + k] = nodev;
  }
}

// ---------------------------------------------------------------------------
// Kernel 2: soft-DTW DP. Forward (alpha) and backward (beta) run as
// independent blocks (dir = blockIdx.x & 1). Edge tile staged in LDS with
// pitch 97 (odd) so both column reads (fwd) and row reads (bwd) are
// bank-conflict-free. Grid: 2*B, Block: 128.
// ---------------------------------------------------------------------------
__global__ __launch_bounds__(128) void gdtw_dp_kernel(
    const float* __restrict__ node, const float* __restrict__ glbL,
    const float* __restrict__ glbU, const float* __restrict__ t2,
    const float* __restrict__ tw,   const float* __restrict__ reg_wt,
    const float* __restrict__ lcl_grad_ub,
    float* __restrict__ alpha, float* __restrict__ beta)
{
  const int b   = blockIdx.x >> 1;
  const int dir = blockIdx.x & 1;
  const int tid = threadIdx.x;

  __shared__ float lbs[MW], ubs[MW], dtws[MW];
  __shared__ float etile[KD * 97];
  __shared__ float abuf[2][KD];
  __shared__ float nrow[KD];
  __shared__ float tcur[KD], tnext[KD];
  __shared__ float red[128];
  __shared__ float T2s;

  float m2 = -1e30f;
  for (int i = tid; i < N2_; i += 128) m2 = fmaxf(m2, t2[b * N2_ + i]);
  red[tid] = m2; __syncthreads();
  for (int s = 64; s > 0; s >>= 1) { if (tid < s) red[tid] = fmaxf(red[tid], red[tid + s]); __syncthreads(); }
  if (tid == 0) T2s = red[0];
  __syncthreads();
  const float T2 = T2s;

  for (int m = tid; m < MW; m += 128) {
    lbs[m] = glbL[b * MW + m] * T2;
    ubs[m] = glbU[b * MW + m] * T2;
  }
  for (int i = tid; i < MW - 1; i += 128) dtws[i] = tw[b * MW + i + 1] - tw[b * MW + i];
  __syncthreads();
  const float rw  = reg_wt[b];
  const float gub = lcl_grad_ub[b];
  const float kstep = 1.f / (float)(KD - 1);

  if (dir == 0) {
    // -------- forward pass: alpha --------
    if (tid < KD) {
      const float v = node[((size_t)b * MW) * KD + tid];
      abuf[0][tid] = v;
      alpha[((size_t)b * MW) * KD + tid] = v;
    }
    __syncthreads();
    int cur = 0;
    for (int i = 0; i < MW - 1; ++i) {
      const float dt = dtws[i], inv_dt = 1.f / dt;
      if (tid < KD) {
        tcur[tid]  = lbs[i]     + (ubs[i]     - lbs[i])     * (tid * kstep);
        tnext[tid] = lbs[i + 1] + (ubs[i + 1] - lbs[i + 1]) * (tid * kstep);
        nrow[tid]  = node[((size_t)b * MW + i + 1) * KD + tid];
      }
      __syncthreads();
      for (int idx = tid; idx < KD * KD; idx += 128) {
        const int k = idx / KD, kp = idx - k * KD;
        const float slope = (tnext[kp] - tcur[k]) * inv_dt;
        const float p0 = fmaxf(-slope, 0.f);
        const float p1 = fmaxf(slope - gub, 0.f);
        const float sm1 = slope - 1.f;
        const float e = rw * sm1 * sm1 * dt + BARRIER_ * (p0 * p0 + p1 * p1);
        etile[k * 97 + kp] = e + abuf[cur][k];
      }
      __syncthreads();
      if (tid < KD) {
        float vmin = 3.4e38f;
        for (int k = 0; k < KD; ++k) vmin = fminf(vmin, etile[k * 97 + tid]);
        float s = 0.f;
        for (int k = 0; k < KD; ++k) s += __expf((vmin - etile[k * 97 + tid]) * INV_GAMMA);
        const float a_new = nrow[tid] + vmin - GAMMA_ * __logf(s);
        abuf[cur ^ 1][tid] = a_new;
        alpha[((size_t)b * MW + i + 1) * KD + tid] = a_new;
      }
      __syncthreads();
      cur ^= 1;
    }
  } else {
    // -------- backward pass: beta --------
    if (tid < KD) {
      abuf[0][tid] = 0.f;
      beta[((size_t)b * MW + MW - 1) * KD + tid] = 0.f;
    }
    __syncthreads();
    int cur = 0;
    for (int i = MW - 2; i >= 0; --i) {
      const float dt = dtws[i], inv_dt = 1.f / dt;
      if (tid < KD) {
        tcur[tid]  = lbs[i]     + (ubs[i]     - lbs[i])     * (tid * kstep);
        tnext[tid] = lbs[i + 1] + (ubs[i + 1] - lbs[i + 1]) * (tid * kstep);
        nrow[tid]  = node[((size_t)b * MW + i + 1) * KD + tid] + abuf[cur][tid];
      }
      __syncthreads();
      for (int idx = tid; idx < KD * KD; idx += 128) {
        const int k = idx / KD, kp = idx - k * KD;
        const float slope = (tnext[kp] - tcur[k]) * inv_dt;
        const float p0 = fmaxf(-slope, 0.f);
        const float p1 = fmaxf(slope - gub, 0.f);
        const float sm1 = slope - 1.f;
        const float e = rw * sm1 * sm1 * dt + BARRIER_ * (p0 * p0 + p1 * p1);
        etile[k * 97 + kp] = e + nrow[kp];
      }
      __syncthreads();
      if (tid < KD) {
        float vmin = 3.4e38f;
        for (int kp = 0; kp < KD; ++kp) vmin = fminf(vmin, etile[tid * 97 + kp]);
        float s = 0.f;
        for (int kp = 0; kp < KD; ++kp) s += __expf((vmin - etile[tid * 97 + kp]) * INV_GAMMA);
        const float b_new = vmin - GAMMA_ * __logf(s);
        abuf[cur ^ 1][tid] = b_new;
        beta[((size_t)b * MW + i) * KD + tid] = b_new;
      }
      __syncthreads();
      cur ^= 1;
    }
  }
}

// ---------------------------------------------------------------------------
// Kernel 3: out[b,m] = sum_k softmax_k(-(alpha+beta)/gamma) * tau
// ---------------------------------------------------------------------------
__global__ __launch_bounds__(256) void gdtw_out_kernel(
    const float* __restrict__ alpha, const float* __restrict__ beta,
    const float* __restrict__ glbL,  const float* __restrict__ glbU,
    const float* __restrict__ t2,    float* __restrict__ out)
{
  const int row = blockIdx.x * 256 + threadIdx.x;
  if (row >= B_ * MW) return;
  const int b = row / MW;
  float T2 = -1e30f;
  for (int i = 0; i < N2_; ++i) T2 = fmaxf(T2, t2[b * N2_ + i]);
  const float lb = glbL[row] * T2;
  const float ub = glbU[row] * T2;
  const float* ar = alpha + (size_t)row * KD;
  const float* br = beta  + (size_t)row * KD;
  float vmin = 3.4e38f;
  for (int k = 0; k < KD; ++k) vmin = fminf(vmin, ar[k] + br[k]);
  float s = 0.f, ws = 0.f;
  const float kstep = 1.f / (float)(KD - 1);
  for (int k = 0; k < KD; ++k) {
    const float p = __expf((vmin - (ar[k] + br[k])) * INV_GAMMA);
    const float tau = lb + (ub - lb) * (k * kstep);
    s += p; ws += p * tau;
  }
  out[row] = ws / s;
}

extern "C" void kernel_launch(void* const* d_in, const int* in_sizes, int n_in,
                              void* d_out, int out_size, void* d_ws, size_t ws_size,
                              hipStream_t stream)
{
  (void)in_sizes; (void)n_in; (void)out_size; (void)ws_size;
  const float* s1f = (const float*)d_in[0];
  const float* s2f = (const float*)d_in[1];
  const float* reg = (const float*)d_in[2];
  const float* gl  = (const float*)d_in[3];
  const float* gu  = (const float*)d_in[4];
  const float* lgu = (const float*)d_in[5];
  const float* t1  = (const float*)d_in[6];
  const float* t2  = (const float*)d_in[7];
  const float* tw  = (const float*)d_in[8];
  float* out = (float*)d_out;

  float* node  = (float*)d_ws;                       // B*MW*KD
  float* alpha = node  + (size_t)B_ * MW * KD;       // B*MW*KD
  float* beta  = alpha + (size_t)B_ * MW * KD;       // B*MW*KD

  dim3 g1(MW / 16, B_);
  gdtw_node_kernel<<<g1, 256, 0, stream>>>(s1f, s2f, gl, gu, t1, t2, tw, node);
  gdtw_dp_kernel<<<2 * B_, 128, 0, stream>>>(node, gl, gu, t2, tw, reg, lgu, alpha, beta);
  gdtw_out_kernel<<<(B_ * MW + 255) / 256, 256, 0, stream>>>(alpha, beta, gl, gu, t2, out);
}